// GMMLoss_84430467104863
// MI455X (gfx1250) — compile-verified
//
#include <hip/hip_runtime.h>
#include <stdint.h>

#define N_G   16
#define DIM   64
#define ROWS  16            // batch rows per workgroup
#define WAVES 4
#define THREADS (WAVES * 32)
#define G_PER_WAVE (N_G / WAVES)

typedef float v2f __attribute__((ext_vector_type(2)));
typedef float v8f __attribute__((ext_vector_type(8)));

__device__ __forceinline__ float fast_ln(float x) {
    return __builtin_amdgcn_logf(x) * 0.69314718055994530942f;   // v_log_f32 = log2
}
__device__ __forceinline__ float fast_exp(float x) {
    return __builtin_amdgcn_exp2f(x * 1.44269504088896340736f);  // v_exp_f32 = exp2
}

__global__ __launch_bounds__(THREADS) void GMMLoss_84430467104863_kernel(
    const float* __restrict__ means, const float* __restrict__ covs,
    const float* __restrict__ weights, const float* __restrict__ targets,
    float* __restrict__ out)
{
    __shared__ __align__(16) float s_t[ROWS * DIM];          // 4 KB targets tile
    __shared__ float s_w[ROWS * N_G];                        // 1 KB weights tile
    __shared__ float s_res[ROWS * N_G];                      // 1 KB quad+logdet results
    // per-wave double-buffered staging: [wave][buf][mu|var][16 rows x 64 f32]
    __shared__ __align__(16) float s_mv[WAVES][2][2][ROWS * DIM];  // 64 KB

    const int tid    = threadIdx.x;
    const int lane   = tid & 31;
    const int wv     = tid >> 5;
    const int m      = lane & 15;      // batch row within tile
    const int khalf  = lane >> 4;      // K-half for WMMA A layout
    const int b0     = blockIdx.x * ROWS;

    // ---- stage targets (256 float4) and weights (256 floats), block-wide ----
    {
        const float4* tg = (const float4*)(targets + (size_t)b0 * DIM);
        float4* st = (float4*)s_t;
        st[tid]           = tg[tid];
        st[tid + THREADS] = tg[tid + THREADS];
        s_w[tid]           = weights[(size_t)b0 * N_G + tid];
        s_w[tid + THREADS] = weights[(size_t)b0 * N_G + tid + THREADS];
    }
    __syncthreads();

    // ---- hoist this lane's target values: identical across all its gaussians ----
    float4 treg[8];
    #pragma unroll
    for (int h = 0; h < 2; ++h) {
        const float4* t4 = (const float4*)(s_t + m * DIM + (2 * khalf + h) * 16);
        #pragma unroll
        for (int j = 0; j < 4; ++j) treg[h * 4 + j] = t4[j];
    }

    const unsigned long long mu_base = (unsigned long long)(uintptr_t)means;
    const unsigned long long va_base = (unsigned long long)(uintptr_t)covs;
    // per-lane 32-bit byte offset into means/covs (GVS addressing: SGPR base + VGPR i32)
    const uint32_t voff0 = (uint32_t)(b0 + khalf) * (uint32_t)(N_G * DIM * 4)
                         + (uint32_t)(m * 16);

    // ---- async stage of one gaussian's 16x64 mu and var tiles into LDS ----
    auto stage_g = [&](int g, int buf) {
        uint32_t vg  = voff0 + (uint32_t)g * (DIM * 4);
        uint32_t lmu = (uint32_t)(uintptr_t)&s_mv[wv][buf][0][0]
                     + (uint32_t)(khalf * DIM * 4 + m * 16);
        uint32_t lva = lmu + (uint32_t)(ROWS * DIM * 4);
        #pragma unroll
        for (int i = 0; i < 8; ++i) {                 // 2 rows per instruction
            asm volatile("global_load_async_to_lds_b128 %0, %1, %2 th:TH_LOAD_NT"
                         :: "v"(lmu), "v"(vg), "s"(mu_base) : "memory");
            asm volatile("global_load_async_to_lds_b128 %0, %1, %2 th:TH_LOAD_NT"
                         :: "v"(lva), "v"(vg), "s"(va_base) : "memory");
            vg  += 2u * N_G * DIM * 4;                // +8192 B (2 global rows)
            lmu += 2u * DIM * 4;                      // +512 B  (2 tile rows)
            lva += 2u * DIM * 4;
        }
    };

    // ---- per-gaussian compute: VALU partials -> one WMMA row-reduction ----
    auto compute = [&](int g, const float* mu, const float* va) {
        v2f a;
        #pragma unroll
        for (int h = 0; h < 2; ++h) {
            int kk = 2 * khalf + h;                   // K-slot (ISA A-matrix layout)
            float acc = 0.0f, prod = 1.0f;
            const float4* m4 = (const float4*)(mu + m * DIM + kk * 16);
            const float4* v4 = (const float4*)(va + m * DIM + kk * 16);
            #pragma unroll
            for (int j = 0; j < 4; ++j) {
                float4 mm = m4[j], vv = v4[j], tt = treg[h * 4 + j];
                float d0 = tt.x - mm.x, d1 = tt.y - mm.y;
                float d2 = tt.z - mm.z, d3 = tt.w - mm.w;
                acc += d0 * d0 * __builtin_amdgcn_rcpf(vv.x);
                acc += d1 * d1 * __builtin_amdgcn_rcpf(vv.y);
                acc += d2 * d2 * __builtin_amdgcn_rcpf(vv.z);
                acc += d3 * d3 * __builtin_amdgcn_rcpf(vv.w);
                prod *= vv.x * vv.y * vv.z * vv.w;    // sum log = log prod (safe range)
            }
            float part = acc + fast_ln(prod);
            if (h == 0) a.x = part; else a.y = part;
        }
        v2f ones; ones.x = 1.0f; ones.y = 1.0f;
        v8f c = {};
        // D[m,n] = sum_k A[m,k] * 1 : full 64-dim reduction per row
        c = __builtin_amdgcn_wmma_f32_16x16x4_f32(false, a, false, ones,
                                                  (short)0, c, false, false);
        if ((lane & 15) == 0) {                       // lane 0 -> rows 0..7, lane 16 -> rows 8..15
            #pragma unroll
            for (int j = 0; j < 8; ++j)
                s_res[(j + 8 * khalf) * N_G + g] = c[j];
        }
    };

    // ---- double-buffered pipeline over this wave's 4 gaussians ----
    const int g0 = wv * G_PER_WAVE;
    stage_g(g0, 0);
    #pragma unroll
    for (int gi = 0; gi < G_PER_WAVE; ++gi) {
        if (gi + 1 < G_PER_WAVE) {
            stage_g(g0 + gi + 1, (gi + 1) & 1);
            asm volatile("s_wait_asynccnt 16" ::: "memory");  // oldest 16 (this buf) done
        } else {
            asm volatile("s_wait_asynccnt 0" ::: "memory");
        }
        compute(g0 + gi, &s_mv[wv][gi & 1][0][0], &s_mv[wv][gi & 1][1][0]);
    }

    __syncthreads();

    // ---- final clipped logsumexp over 16 gaussians, one lane per row ----
    if (tid < ROWS) {
        float lp[N_G];
        float mx = -3.0e38f;
        #pragma unroll
        for (int g = 0; g < N_G; ++g) {
            float s = s_res[tid * N_G + g];
            float l = -0.5f * (117.62413225019811f + s);   // 64*log(2*pi) + logdet + quad
            l = fminf(fmaxf(l, -100.0f), 0.0f);
            l += fast_ln(s_w[tid * N_G + g]);
            lp[g] = l;
            mx = fmaxf(mx, l);
        }
        float sum = 0.0f;
        #pragma unroll
        for (int g = 0; g < N_G; ++g) sum += fast_exp(lp[g] - mx);
        out[b0 + tid] = -(mx + fast_ln(sum));
    }
}

extern "C" void kernel_launch(void* const* d_in, const int* in_sizes, int n_in,
                              void* d_out, int out_size, void* d_ws, size_t ws_size,
                              hipStream_t stream) {
    const float* means   = (const float*)d_in[0];
    const float* covs    = (const float*)d_in[1];
    const float* weights = (const float*)d_in[2];
    const float* targets = (const float*)d_in[3];
    float* out = (float*)d_out;

    const int B = in_sizes[3] / DIM;   // targets is B x 64
    dim3 grid(B / ROWS), block(THREADS);
    GMMLoss_84430467104863_kernel<<<grid, block, 0, stream>>>(means, covs, weights, targets, out);
}